// HungarianMatcher_86912958202054
// MI455X (gfx1250) — compile-verified
//
#include <hip/hip_runtime.h>
#include <hip/hip_bf16.h>
#include <stdint.h>

// Problem constants (match reference).
constexpr int   kB = 128, kN = 900, kC = 91, kM = 100;
constexpr int   kTN = 36;                 // n-rows per block; 900 = 25 * 36
constexpr int   kRS = 9;                  // box record stride (dwords), odd => <=2-way LDS conflicts
constexpr float kAlpha = 0.25f;
constexpr float kWClass = 2.0f, kWBbox = 5.0f, kWGiou = 2.0f;
constexpr float kEps = 1e-8f;

// Low 32 bits of a flat shared-aperture pointer are the wave-relative LDS byte address.
__device__ __forceinline__ uint32_t lds_addr_of(const void* p) {
  return (uint32_t)(uintptr_t)p;
}

// CDNA5 async DMA: copies from the global address saddr + voff to the LDS address in lds  (GVS mode, tracked by ASYNCcnt).
__device__ __forceinline__ void async_ld_b32(uint32_t lds, const void* saddr, uint32_t voff) {
  asm volatile("global_load_async_to_lds_b32 %0, %1, %2"
               :: "v"(lds), "v"(voff), "s"(saddr) : "memory");
}
__device__ __forceinline__ void async_ld_b128(uint32_t lds, const void* saddr, uint32_t voff) {
  asm volatile("global_load_async_to_lds_b128 %0, %1, %2"
               :: "v"(lds), "v"(voff), "s"(saddr) : "memory");
}
__device__ __forceinline__ void wait_async0() {
  asm volatile("s_wait_asynccnt 0" ::: "memory");
}

__global__ __launch_bounds__(256)
void hungarian_cost_kernel(const float* __restrict__ logits,   // [B,N,C]
                           const float* __restrict__ pboxes,   // [B,N,4] cxcywh
                           const int*   __restrict__ labels,   // [B,M]
                           const float* __restrict__ tboxes,   // [B,M,4] cxcywh
                           float* __restrict__ out) {          // [B,N,M]
  // Box record layout (stride kRS=9 dwords):
  //  [0..3] cx,cy,w,h   [4..7] x0,y0,x1,y1   [8] area
  __shared__ __align__(16) float sCls[kTN * kC];   // logits tile -> focal cost (in place)
  __shared__ float sT[kM * kRS];                   // target box records
  __shared__ float sP[kTN * kRS];                  // pred box records
  __shared__ __align__(16) int sLab[kM];

  const int tid = threadIdx.x;
  const int b  = blockIdx.y;
  const int n0 = blockIdx.x * kTN;

  const float* glog = logits + ((size_t)b * kN + n0) * kC;   // contiguous 36x91 tile,
  const float* gpb  = pboxes + ((size_t)b * kN + n0) * 4;    // tile byte size 13104 = 819*16
  const float* gtb  = tboxes + (size_t)b * kM * 4;
  const int*   glab = labels + (size_t)b * kM;

  const uint32_t aCls = lds_addr_of(sCls);
  const uint32_t aT   = lds_addr_of(sT);
  const uint32_t aP   = lds_addr_of(sP);
  const uint32_t aL   = lds_addr_of(sLab);

  // ---- Phase 1: async DMA global -> LDS (no VGPR round trip) ----
  // Logits tile: 819 x b128 (16B-aligned both sides, linear layout).
  #pragma unroll 1
  for (int i = tid; i < (kTN * kC) / 4; i += 256)
    async_ld_b128(aCls + 16u * (uint32_t)i, glog, 16u * (uint32_t)i);

  #pragma unroll 1
  for (int i = tid; i < kM * 4; i += 256) {                 // target boxes -> records
    int m = i >> 2, k = i & 3;
    async_ld_b32(aT + 4u * (uint32_t)(m * kRS + k), gtb, 4u * (uint32_t)i);
  }
  if (tid < kTN * 4) {                                      // pred boxes -> records
    int r = tid >> 2, k = tid & 3;
    async_ld_b32(aP + 4u * (uint32_t)(r * kRS + k), gpb, 4u * (uint32_t)tid);
  }
  if (tid < kM / 4)                                         // labels: 25 x b128
    async_ld_b128(aL + 16u * (uint32_t)tid, glab, 16u * (uint32_t)tid);

  wait_async0();        // each wave drains its own ASYNCcnt
  __syncthreads();      // then all waves' DMA results are visible in LDS

  // ---- Phase 2a: focal class cost, in place over the logits tile ----
  #pragma unroll 1
  for (int i = tid; i < kTN * kC; i += 256) {
    float x  = sCls[i];
    float s  = __frcp_rn(1.0f + __expf(-x));                // sigmoid
    float om = 1.0f - s;
    float neg = (1.0f - kAlpha) * (s * s)   * (-__logf(om + kEps));
    float pos = kAlpha          * (om * om) * (-__logf(s  + kEps));
    sCls[i] = pos - neg;
  }
  // ---- Phase 2b: cxcywh -> xyxy + area (hoisted out of the NxM loop) ----
  if (tid < kM + kTN) {
    float* r = (tid < kM) ? &sT[tid * kRS] : &sP[(tid - kM) * kRS];
    float cx = r[0], cy = r[1], w = r[2], h = r[3];
    float x0 = cx - 0.5f * w, y0 = cy - 0.5f * h;
    float x1 = cx + 0.5f * w, y1 = cy + 0.5f * h;
    r[4] = x0; r[5] = y0; r[6] = x1; r[7] = y1;
    r[8] = (x1 - x0) * (y1 - y0);
  }
  __syncthreads();

  // ---- Phase 3: each thread emits a float4 of 4 consecutive m (100 = 25 quads) ----
  #pragma unroll 1
  for (int idx = tid; idx < kTN * (kM / 4); idx += 256) {
    int n  = idx / (kM / 4);
    int m0 = (idx - n * (kM / 4)) * 4;

    // Pred record: loaded once, reused for 4 outputs.
    const float* pr = &sP[n * kRS];
    float pcx = pr[0], pcy = pr[1], pw = pr[2], ph = pr[3];
    float px0 = pr[4], py0 = pr[5], px1 = pr[6], py1 = pr[7];
    float pa  = pr[8];

    float4 res;
    float* resv = &res.x;
    #pragma unroll
    for (int j = 0; j < 4; ++j) {
      int m = m0 + j;
      const float* tr = &sT[m * kRS];

      float cc = sCls[n * kC + sLab[m]];                    // class-cost gather

      float cb = fabsf(pcx - tr[0]) + fabsf(pcy - tr[1])    // L1 on cxcywh
               + fabsf(pw  - tr[2]) + fabsf(ph  - tr[3]);

      float ix0 = fmaxf(px0, tr[4]), iy0 = fmaxf(py0, tr[5]);
      float ix1 = fminf(px1, tr[6]), iy1 = fminf(py1, tr[7]);
      float iw = fmaxf(ix1 - ix0, 0.0f), ih = fmaxf(iy1 - iy0, 0.0f);
      float inter = iw * ih;
      float uni = pa + tr[8] - inter;

      float ex0 = fminf(px0, tr[4]), ey0 = fminf(py0, tr[5]);
      float ex1 = fmaxf(px1, tr[6]), ey1 = fmaxf(py1, tr[7]);
      float ew = fmaxf(ex1 - ex0, 0.0f), eh = fmaxf(ey1 - ey0, 0.0f);
      float ea = ew * eh;

      // giou = iou - (ea - uni)/ea ; cost adds -W_GIOU * giou
      float giou = __fdividef(inter, uni) - __fdividef(ea - uni, ea);

      resv[j] = kWClass * cc + kWBbox * cb - kWGiou * giou;
    }

    // Coalesced 16B store (rows are 400 B, quad-aligned).
    float* orow = out + ((size_t)b * kN + (size_t)(n0 + n)) * (size_t)kM + (size_t)m0;
    *reinterpret_cast<float4*>(orow) = res;
  }
}

extern "C" void kernel_launch(void* const* d_in, const int* in_sizes, int n_in,
                              void* d_out, int out_size, void* d_ws, size_t ws_size,
                              hipStream_t stream) {
  const float* logits = (const float*)d_in[0];   // [B,N,C] f32
  const float* pboxes = (const float*)d_in[1];   // [B,N,4] f32
  const int*   labels = (const int*)d_in[2];     // [B,M] i32
  const float* tboxes = (const float*)d_in[3];   // [B,M,4] f32
  float* out = (float*)d_out;                    // [B,N,M] f32

  dim3 grid(kN / kTN, kB);   // 25 x 128 blocks
  dim3 block(256);           // 8 wave32s
  hipLaunchKernelGGL(hungarian_cost_kernel, grid, block, 0, stream,
                     logits, pboxes, labels, tboxes, out);
  (void)in_sizes; (void)n_in; (void)out_size; (void)d_ws; (void)ws_size;
}